// SampleAndAggregate_71725953843785
// MI455X (gfx1250) — compile-verified
//
#include <hip/hip_runtime.h>
#include <cstdint>

typedef float v2f __attribute__((ext_vector_type(2)));
typedef float v8f __attribute__((ext_vector_type(8)));

// ---------------- Threefry-2x32 (matches JAX lax_prng exactly) ----------------
__host__ __device__ inline uint32_t rotl32(uint32_t v, uint32_t n) {
  return (v << n) | (v >> (32u - n));
}
__host__ __device__ inline void threefry2x32(uint32_t k0, uint32_t k1,
                                             uint32_t x0, uint32_t x1,
                                             uint32_t& o0, uint32_t& o1) {
  const uint32_t ks2 = k0 ^ k1 ^ 0x1BD11BDAu;
  x0 += k0; x1 += k1;
#define TF_R(r) { x0 += x1; x1 = rotl32(x1, r); x1 ^= x0; }
  TF_R(13u) TF_R(15u) TF_R(26u) TF_R(6u)  x0 += k1;  x1 += ks2 + 1u;
  TF_R(17u) TF_R(29u) TF_R(16u) TF_R(24u) x0 += ks2; x1 += k0 + 2u;
  TF_R(13u) TF_R(15u) TF_R(26u) TF_R(6u)  x0 += k0;  x1 += k1 + 3u;
  TF_R(17u) TF_R(29u) TF_R(16u) TF_R(24u) x0 += k1;  x1 += ks2 + 4u;
  TF_R(13u) TF_R(15u) TF_R(26u) TF_R(6u)  x0 += ks2; x1 += k0 + 5u;
#undef TF_R
  o0 = x0; o1 = x1;
}

// -------- Sampling: out[g] = adj[ids[g/ns]*128 + (bits(g) & 127)] --------
// JAX pairing: counts iota(n) split in halves, pair (t, t+n/2) -> (word0, word1).
__global__ void sage_sample_kernel(const int* __restrict__ ids,
                                   const int* __restrict__ adj,
                                   int* __restrict__ out,
                                   int M, int ns, uint32_t kk0, uint32_t kk1) {
  int t = blockIdx.x * blockDim.x + threadIdx.x;
  int half = (M * ns) >> 1;
  if (t >= half) return;
  uint32_t o0, o1;
  threefry2x32(kk0, kk1, (uint32_t)t, (uint32_t)(t + half), o0, o1);
  int g0 = t, g1 = t + half;
  out[g0] = adj[(size_t)ids[g0 / ns] * 128 + (int)(o0 & 127u)];
  out[g1] = adj[(size_t)ids[g1 / ns] * 128 + (int)(o1 & 127u)];
}

// -------- Group mean: out[i,:] = mean_j src[idx ? idx[i*ns+j] : i*ns+j, :] ----
__global__ void sage_mean_kernel(const float* __restrict__ src,
                                 const int* __restrict__ idx,
                                 float* __restrict__ out,
                                 int M, int ns, float inv) {
  int t = blockIdx.x * blockDim.x + threadIdx.x;
  int total = M << 6;               // M * 64 float4 groups (256 cols)
  if (t >= total) return;
  int row = t >> 6, c4 = t & 63;
  float sx = 0.f, sy = 0.f, sz = 0.f, sw = 0.f;
  int gbase = row * ns;
  for (int j = 0; j < ns; ++j) {
    int g = gbase + j;
    long sr = idx ? (long)idx[g] : (long)g;
    const float4 v = *reinterpret_cast<const float4*>(src + sr * 256 + c4 * 4);
    sx += v.x; sy += v.y; sz += v.z; sw += v.w;
  }
  float4 o; o.x = sx * inv; o.y = sy * inv; o.z = sz * inv; o.w = sw * inv;
  *reinterpret_cast<float4*>(out + (size_t)row * 256 + c4 * 4) = o;
}

// -------- WMMA f32 GEMM + relu: Out[m, colOff + n] = relu(A[m,:256] @ W[:,n]) --
// A rows optionally gathered through Aidx. W is [256,128] row-major (K x N).
// Block = 128 threads = 4 waves: 2 row tiles x 2 col halves; the 32x128 weight
// chunk is staged once per block into LDS (double buffered, k-pair interleaved
// so each B fragment is a single ds_load_b64). Grids are exact: no wave guard,
// so __syncthreads is uniform and EXEC is all-ones at every WMMA.
__global__ void sage_wmma_gemm_relu(const float* __restrict__ A,
                                    const int* __restrict__ Aidx,
                                    int M, const float* __restrict__ W,
                                    float* __restrict__ Out, int colOff) {
  __shared__ float lw[2][32 * 128];          // 2 x 16KB chunks

  const int tid  = threadIdx.x;              // 0..127
  const int lane = tid & 31;
  const int wv   = blockIdx.x * 4 + (tid >> 5);
  const int rowTile = wv >> 1;
  const int halfSel = wv & 1;
  const int m0   = rowTile << 4;
  const int r    = lane & 15;
  const int khi  = (lane >> 4) << 1;         // 0 for lanes 0-15, 2 for 16-31
  const int nCol = lane & 15;
  const int nLoc0 = halfSel * 64 + nCol;     // LDS-local column

  long arow = Aidx ? (long)Aidx[m0 + r] : (long)(m0 + r);
  const float* ap = A + arow * 256 + khi;

  // Stage one 32x128 chunk of W into lw[buf], interleaving k pairs:
  // element (k, n) -> lw[buf][((k>>1)*128 + n)*2 + (k&1)]
  auto stage = [&](int buf, int kb0) {
    const float4* Wp = reinterpret_cast<const float4*>(W + (size_t)kb0 * 128);
#pragma unroll
    for (int i = 0; i < 8; ++i) {
      int f  = tid + i * 128;                // float4 index in chunk, 0..1023
      int k  = f >> 5;                       // 0..31
      int n  = (f & 31) * 4;
      float4 v = Wp[f];
      float* d = &lw[buf][((k >> 1) * 128 + n) * 2 + (k & 1)];
      d[0] = v.x; d[2] = v.y; d[4] = v.z; d[6] = v.w;
    }
  };

  v8f acc0 = {}, acc1 = {}, acc2 = {}, acc3 = {};
  stage(0, 0);
#pragma unroll 1
  for (int c = 0; c < 8; ++c) {
    __syncthreads();
    if (c < 7) stage((c + 1) & 1, (c + 1) * 32);
    const int buf = c & 1;
    const int kb0 = c * 32;
#pragma unroll
    for (int kk = 0; kk < 8; ++kk) {
      const float2 av = *reinterpret_cast<const float2*>(ap + kb0 + kk * 4);
      v2f a; a.x = av.x; a.y = av.y;
      const int pi = kk * 2 + (lane >> 4);   // k-pair index within chunk
      const float* lb = &lw[buf][(pi * 128 + nLoc0) * 2];
      v2f b0 = *reinterpret_cast<const v2f*>(lb + 0);
      v2f b1 = *reinterpret_cast<const v2f*>(lb + 32);
      v2f b2 = *reinterpret_cast<const v2f*>(lb + 64);
      v2f b3 = *reinterpret_cast<const v2f*>(lb + 96);
      acc0 = __builtin_amdgcn_wmma_f32_16x16x4_f32(false, a, false, b0, (short)0, acc0, false, false);
      acc1 = __builtin_amdgcn_wmma_f32_16x16x4_f32(false, a, false, b1, (short)0, acc1, false, false);
      acc2 = __builtin_amdgcn_wmma_f32_16x16x4_f32(false, a, false, b2, (short)0, acc2, false, false);
      acc3 = __builtin_amdgcn_wmma_f32_16x16x4_f32(false, a, false, b3, (short)0, acc3, false, false);
    }
  }

  // C/D layout: VGPR v -> row v (lanes 0-15) / row v+8 (lanes 16-31), col = lane&15
  const int rAdd  = (lane >> 4) << 3;
  const int nBase = colOff + halfSel * 64 + nCol;
#pragma unroll
  for (int v = 0; v < 8; ++v) {
    int row = m0 + v + rAdd;
    float* op = Out + (size_t)row * 256 + nBase;
    op[0]  = fmaxf(acc0[v], 0.f);
    op[16] = fmaxf(acc1[v], 0.f);
    op[32] = fmaxf(acc2[v], 0.f);
    op[48] = fmaxf(acc3[v], 0.f);
  }
}

// -------- Finalize: out[i] = dot(h_i, w) / max(||h_i||, eps) + b --------
__global__ void sage_finalize_kernel(const float* __restrict__ H,
                                     const float* __restrict__ wo,
                                     const float* __restrict__ bo,
                                     float* __restrict__ out, int M) {
  int wv   = (blockIdx.x * blockDim.x + threadIdx.x) >> 5;
  int lane = threadIdx.x & 31;
  if (wv >= M) return;
  const float* hp = H + (size_t)wv * 256 + lane * 8;
  const float* wp = wo + lane * 8;
  float s2 = 0.f, d = 0.f;
  for (int i = 0; i < 8; ++i) { float h = hp[i]; s2 += h * h; d += h * wp[i]; }
  for (int m = 16; m >= 1; m >>= 1) {
    s2 += __shfl_xor(s2, m, 32);
    d  += __shfl_xor(d,  m, 32);
  }
  if (lane == 0) out[wv] = d * rsqrtf(fmaxf(s2, 1e-12f)) + bo[0];
}

extern "C" void kernel_launch(void* const* d_in, const int* in_sizes, int n_in,
                              void* d_out, int out_size, void* d_ws, size_t ws_size,
                              hipStream_t stream) {
  const float* features  = (const float*)d_in[0];   // [100000,256]
  const float* w_self_0  = (const float*)d_in[1];   // [256,128]
  const float* w_neigh_0 = (const float*)d_in[2];   // [256,128]
  const float* w_self_1  = (const float*)d_in[3];   // [256,128]
  const float* w_neigh_1 = (const float*)d_in[4];   // [256,128]
  const float* w_out     = (const float*)d_in[5];   // [256,1]
  const float* b_out     = (const float*)d_in[6];   // [1]
  const int*   node_ids  = (const int*)d_in[7];     // [1024]
  const int*   adj       = (const int*)d_in[8];     // [100000,128]
  float* out = (float*)d_out;

  const int BATCH = 1024, M1 = 10240, M2 = 256000;

  // Workspace carve-up (all sizes multiples of 256 B)
  char* ws = (char*)d_ws;
  int*   s1  = (int*)ws;   ws += (size_t)M1 * sizeof(int);
  int*   s2  = (int*)ws;   ws += (size_t)M2 * sizeof(int);
  float* X0n = (float*)ws; ws += (size_t)BATCH * 256 * sizeof(float);
  float* X1n = (float*)ws; ws += (size_t)M1 * 256 * sizeof(float);
  float* H10 = (float*)ws; ws += (size_t)BATCH * 256 * sizeof(float);
  float* H11 = (float*)ws; ws += (size_t)M1 * 256 * sizeof(float);
  float* X2n = (float*)ws; ws += (size_t)BATCH * 256 * sizeof(float);
  float* H2  = (float*)ws; ws += (size_t)BATCH * 256 * sizeof(float);

  // Host-side key schedule: key(42) -> fold_in(k) -> split -> lower-bits key k2
  uint32_t hop_k0[2], hop_k1[2];
  for (int k = 0; k < 2; ++k) {
    uint32_t f0, f1, a0, b0, a1, b1;
    threefry2x32(0u, 42u, 0u, (uint32_t)k, f0, f1);   // fold_in
    threefry2x32(f0, f1, 0u, 2u, a0, b0);             // split: pair (0,2)
    threefry2x32(f0, f1, 1u, 3u, a1, b1);             // split: pair (1,3)
    hop_k0[k] = b0; hop_k1[k] = b1;                   // k2 = [b0, b1]
  }

  // 1) sampling chain
  sage_sample_kernel<<<(BATCH * 10 / 2 + 255) / 256, 256, 0, stream>>>(
      node_ids, adj, s1, BATCH, 10, hop_k0[0], hop_k1[0]);
  sage_sample_kernel<<<(M1 * 25 / 2 + 255) / 256, 256, 0, stream>>>(
      s1, adj, s2, M1, 25, hop_k0[1], hop_k1[1]);

  // 2) neighbor means from the feature table (gathers stay in L2: table = 102 MB)
  sage_mean_kernel<<<((M1 << 6) + 255) / 256, 256, 0, stream>>>(
      features, s2, X1n, M1, 25, 1.f / 25.f);
  sage_mean_kernel<<<((BATCH << 6) + 255) / 256, 256, 0, stream>>>(
      features, s1, X0n, BATCH, 10, 0.1f);

  // 3) layer 0: H = relu([self@Ws | mean@Wn])
  sage_wmma_gemm_relu<<<(M1 / 16) * 2 / 4, 128, 0, stream>>>(
      features, s1, M1, w_self_0, H11, 0);
  sage_wmma_gemm_relu<<<(M1 / 16) * 2 / 4, 128, 0, stream>>>(
      X1n, nullptr, M1, w_neigh_0, H11, 128);
  sage_wmma_gemm_relu<<<(BATCH / 16) * 2 / 4, 128, 0, stream>>>(
      features, node_ids, BATCH, w_self_0, H10, 0);
  sage_wmma_gemm_relu<<<(BATCH / 16) * 2 / 4, 128, 0, stream>>>(
      X0n, nullptr, BATCH, w_neigh_0, H10, 128);

  // 4) layer 1 input mean over consecutive groups of 10 rows of H11
  sage_mean_kernel<<<((BATCH << 6) + 255) / 256, 256, 0, stream>>>(
      H11, nullptr, X2n, BATCH, 10, 0.1f);

  // 5) layer 1 GEMMs
  sage_wmma_gemm_relu<<<(BATCH / 16) * 2 / 4, 128, 0, stream>>>(
      H10, nullptr, BATCH, w_self_1, H2, 0);
  sage_wmma_gemm_relu<<<(BATCH / 16) * 2 / 4, 128, 0, stream>>>(
      X2n, nullptr, BATCH, w_neigh_1, H2, 128);

  // 6) l2-normalize + Dense(1)
  sage_finalize_kernel<<<BATCH / 4, 128, 0, stream>>>(H2, w_out, b_out, out, BATCH);
}